// STAR_65085934403759
// MI455X (gfx1250) — compile-verified
//
#include <hip/hip_runtime.h>
#include <hip/hip_bf16.h>

typedef __attribute__((ext_vector_type(2))) float v2f;
typedef __attribute__((ext_vector_type(8))) float v8f;

// Only check on the DEVICE pass: target builtins are invisible to the host pass.
#if defined(__HIP_DEVICE_COMPILE__) && !__has_builtin(__builtin_amdgcn_wmma_f32_16x16x4_f32)
#error "builtin_amdgcn_wmma_f32_16x16x4_f32 not available on gfx1250 device pass"
#endif

#define GTM 64
#define GTN 128
#define GTK 16
#define DD  1024   // 2W
#define WW  512
#define NPREV 8

// ---------------------------------------------------------------------------
// Transpose 1024x1024 (W_d -> WdT) so every GEMM operand is row-major natural.
// ---------------------------------------------------------------------------
__global__ __launch_bounds__(256) void STAR_transpose_1024(
    const float* __restrict__ in, float* __restrict__ out) {
  __shared__ float tile[32][33];
  const int bx = blockIdx.x * 32, by = blockIdx.y * 32;
  const int tx = threadIdx.x, ty = threadIdx.y;  // 32 x 8
#pragma unroll
  for (int r = 0; r < 32; r += 8)
    tile[ty + r][tx] = in[(size_t)(by + ty + r) * DD + bx + tx];
  __syncthreads();
#pragma unroll
  for (int r = 0; r < 32; r += 8)
    out[(size_t)(bx + ty + r) * DD + by + tx] = tile[tx][ty + r];
}

// ---------------------------------------------------------------------------
// Generic fp32 WMMA GEMM:  C[MxN] = alpha*(A[MxK] @ B[KxN]) (+ addC) (+ I)
// K = N = 1024 fixed; M via grid.y. WG = 256 thr = 8 waves (4 along M, 2
// along N); each wave computes 16x64 with 4 v8f accumulators using
// v_wmma_f32_16x16x4_f32. Tiles staged in LDS (A pad->stride 17, B pad->
// stride 144 so the two 16-lane halves hit disjoint bank ranges).
// ---------------------------------------------------------------------------
__global__ __launch_bounds__(256) void STAR_gemm_wmma(
    const float* __restrict__ A, const float* __restrict__ B,
    float* __restrict__ C, const float* __restrict__ addC,
    float alpha, int addIdent) {
  __shared__ float As[GTM][GTK + 1];    // 64 x 17
  __shared__ float Bs[GTK][GTN + 16];   // 16 x 144
  const int tid  = threadIdx.x;
  const int lane = tid & 31;
  const int wid  = tid >> 5;
  const int wm   = wid & 3;    // wave row   (4)
  const int wn   = wid >> 2;   // wave col   (2)
  const int half = lane >> 4;  // 0: lanes 0-15, 1: lanes 16-31
  const int l16  = lane & 15;
  const int m0 = blockIdx.y * GTM;
  const int n0 = blockIdx.x * GTN;

  // tile-load assignments (dims are exact multiples of the tile: no guards)
  const int arow = tid >> 2;          // 0..63
  const int acol = (tid & 3) << 2;    // 0,4,8,12
  const int brow = tid >> 5;          // 0..7 (+8)
  const int bcol = (tid & 31) << 2;   // 0..124

  v8f acc[4] = {};

  for (int k0 = 0; k0 < DD; k0 += GTK) {
    const float4 av  = *(const float4*)(A + (size_t)(m0 + arow) * DD + k0 + acol);
    const float4 bv0 = *(const float4*)(B + (size_t)(k0 + brow) * DD + n0 + bcol);
    const float4 bv1 = *(const float4*)(B + (size_t)(k0 + brow + 8) * DD + n0 + bcol);
    As[arow][acol + 0] = av.x;  As[arow][acol + 1] = av.y;
    As[arow][acol + 2] = av.z;  As[arow][acol + 3] = av.w;
    Bs[brow][bcol + 0] = bv0.x; Bs[brow][bcol + 1] = bv0.y;
    Bs[brow][bcol + 2] = bv0.z; Bs[brow][bcol + 3] = bv0.w;
    Bs[brow + 8][bcol + 0] = bv1.x; Bs[brow + 8][bcol + 1] = bv1.y;
    Bs[brow + 8][bcol + 2] = bv1.z; Bs[brow + 8][bcol + 3] = bv1.w;
    __syncthreads();

#pragma unroll
    for (int kk = 0; kk < GTK; kk += 4) {
      // A frag (16x4 f32): lanes 0-15 -> K=kk,kk+1 ; lanes 16-31 -> K=kk+2,kk+3
      v2f a;
      a.x = As[wm * 16 + l16][kk + 2 * half + 0];
      a.y = As[wm * 16 + l16][kk + 2 * half + 1];
#pragma unroll
      for (int j = 0; j < 4; ++j) {
        v2f b;  // B frag (4x16 f32), same K split across halves
        b.x = Bs[kk + 2 * half + 0][wn * 64 + j * 16 + l16];
        b.y = Bs[kk + 2 * half + 1][wn * 64 + j * 16 + l16];
        acc[j] = __builtin_amdgcn_wmma_f32_16x16x4_f32(
            false, a, false, b, (short)0, acc[j], false, false);
      }
    }
    __syncthreads();
  }

  // epilogue: D layout -> VGPR r: lanes 0-15 M=r, lanes 16-31 M=r+8; N=l16
#pragma unroll
  for (int j = 0; j < 4; ++j) {
    const int n = n0 + wn * 64 + j * 16 + l16;
#pragma unroll
    for (int r = 0; r < 8; ++r) {
      const int m = m0 + wm * 16 + r + 8 * half;
      float v = alpha * acc[j][r];
      if (addC) v += addC[(size_t)m * DD + n];
      if (addIdent && (m == n)) v += 1.0f;
      C[(size_t)m * DD + n] = v;
    }
  }
}

// ---------------------------------------------------------------------------
// Hard threshold: keep top-32 |v| per 1024-row. One wave per row; 32 values
// per lane in VGPRs; 31-step bitwise binary search on |v| bit pattern
// (monotone for non-negative floats); wave32 count via __shfl_xor.
// ---------------------------------------------------------------------------
__global__ __launch_bounds__(256) void STAR_hard_thr(
    const float* __restrict__ in, float* __restrict__ out) {
  const int row  = blockIdx.x * 8 + (threadIdx.x >> 5);
  const int lane = threadIdx.x & 31;
  const float* r = in + (size_t)row * DD;
  float v[32]; unsigned u[32];
#pragma unroll
  for (int j = 0; j < 32; ++j) {
    v[j] = r[j * 32 + lane];
    u[j] = __float_as_uint(v[j]) & 0x7fffffffu;
  }
  unsigned thr = 0u;
  for (int bit = 30; bit >= 0; --bit) {
    const unsigned cand = thr | (1u << bit);
    int c = 0;
#pragma unroll
    for (int j = 0; j < 32; ++j) c += (u[j] >= cand) ? 1 : 0;
#pragma unroll
    for (int off = 16; off > 0; off >>= 1) c += __shfl_xor(c, off, 32);
    if (c >= 32) thr = cand;   // uniform across the wave
  }
  float* o = out + (size_t)row * DD;
#pragma unroll
  for (int j = 0; j < 32; ++j)
    o[j * 32 + lane] = (u[j] >= thr) ? v[j] : 0.0f;
}

// ---------------------------------------------------------------------------
// Zero the small reduction buffers (mD 512 + mDp 8*512).
// ---------------------------------------------------------------------------
__global__ void STAR_zero_small(float* __restrict__ p) {
  const int t = blockIdx.x * 256 + threadIdx.x;
  if (t < WW + NPREV * WW) p[t] = 0.0f;
}

// ---------------------------------------------------------------------------
// Power spectrum reduce: dst[plane][w] += sum_b (z[b][w]^2 + z[b][w+512]^2)
// grid.x = 64 batch chunks (32 rows each), grid.y = planes.
// ---------------------------------------------------------------------------
__global__ __launch_bounds__(256) void STAR_reduce_power(
    const float* __restrict__ src, float* __restrict__ dst) {
  const float* z = src + (size_t)blockIdx.y * 2048 * DD + (size_t)blockIdx.x * 32 * DD;
  float* o = dst + (size_t)blockIdx.y * WW;
  const int t = threadIdx.x;
  float a0 = 0.0f, a1 = 0.0f;
  for (int r = 0; r < 32; ++r) {
    const float* row = z + (size_t)r * DD;
    const float x0 = row[t],        y0 = row[t + WW];
    const float x1 = row[t + 256],  y1 = row[t + 256 + WW];
    a0 += x0 * x0 + y0 * y0;
    a1 += x1 * x1 + y1 * y1;
  }
  atomicAdd(&o[t], a0);
  atomicAdd(&o[t + 256], a1);
}

// ---------------------------------------------------------------------------
// Epilogue: normalizations + single-head attention + gating, one WG of 512.
// ---------------------------------------------------------------------------
__global__ __launch_bounds__(512) void STAR_finalize(
    const float* __restrict__ mD, const float* __restrict__ mDp,
    const float* __restrict__ in_w, const float* __restrict__ in_b,
    const float* __restrict__ out_w, const float* __restrict__ out_b,
    const float* __restrict__ Wm, const float* __restrict__ Wa,
    const float* __restrict__ ba, float* __restrict__ outv) {
  __shared__ float red[WW];
  __shared__ float mDn[WW];
  __shared__ float mDpn[NPREV][WW];
  __shared__ float qv[WW];
  __shared__ float kv[NPREV][WW];
  __shared__ float vvv[NPREV][WW];
  __shared__ float attn[WW];
  __shared__ float ao[WW];
  __shared__ float aw[NPREV];
  const int t = threadIdx.x;

  auto tree_min = [&](float x) -> float {
    red[t] = x; __syncthreads();
    for (int s = 256; s > 0; s >>= 1) {
      if (t < s) red[t] = fminf(red[t], red[t + s]);
      __syncthreads();
    }
    const float r0 = red[0]; __syncthreads();
    return r0;
  };
  auto tree_max = [&](float x) -> float {
    red[t] = x; __syncthreads();
    for (int s = 256; s > 0; s >>= 1) {
      if (t < s) red[t] = fmaxf(red[t], red[t + s]);
      __syncthreads();
    }
    const float r0 = red[0]; __syncthreads();
    return r0;
  };

  {  // mD_n
    const float x = mD[t];
    const float mn = tree_min(x);
    const float mx = tree_max(x);
    mDn[t] = (x - mn) / (mx - mn + 1e-8f);
  }
  for (int p = 0; p < NPREV; ++p) {  // mDp_n, per-row min/max
    const float y = mDp[p * WW + t];
    const float mn = tree_min(y);
    const float mx = tree_max(y);
    mDpn[p][t] = (y - mn) / (mx - mn + 1e-8f);
  }
  __syncthreads();

  {  // q = mD_n @ in_proj_w[:W].T + b
    const float* wr = in_w + (size_t)t * WW;
    float s = 0.0f;
    for (int w = 0; w < WW; ++w) s += wr[w] * mDn[w];
    qv[t] = s + in_b[t];
  }
  for (int p = 0; p < NPREV; ++p) {  // k, v
    const float* wk = in_w + (size_t)(WW + t) * WW;
    const float* wv = in_w + (size_t)(2 * WW + t) * WW;
    float sk = 0.0f, sv = 0.0f;
    for (int w = 0; w < WW; ++w) {
      const float m = mDpn[p][w];
      sk += wk[w] * m; sv += wv[w] * m;
    }
    kv[p][t]  = sk + in_b[WW + t];
    vvv[p][t] = sv + in_b[2 * WW + t];
  }
  __syncthreads();

  if (t < NPREV) {  // scores
    float s = 0.0f;
    for (int i = 0; i < WW; ++i) s += kv[t][i] * qv[i];
    red[t] = s * 0.04419417382415922f;  // 1/sqrt(512)
  }
  __syncthreads();
  if (t == 0) {  // softmax over 8
    float m = red[0];
    for (int p = 1; p < NPREV; ++p) m = fmaxf(m, red[p]);
    float e[NPREV]; float den = 0.0f;
    for (int p = 0; p < NPREV; ++p) { e[p] = expf(red[p] - m); den += e[p]; }
    for (int p = 0; p < NPREV; ++p) aw[p] = e[p] / den;
  }
  __syncthreads();

  {  // attn = attn_w @ v
    float s = 0.0f;
    for (int p = 0; p < NPREV; ++p) s += aw[p] * vvv[p][t];
    attn[t] = s;
  }
  __syncthreads();
  {  // attn_out = attn @ out_proj_w.T + b
    const float* wr = out_w + (size_t)t * WW;
    float s = 0.0f;
    for (int w = 0; w < WW; ++w) s += wr[w] * attn[w];
    ao[t] = s + out_b[t];
  }
  __syncthreads();

  float multv, addv;
  {
    const float* wm = Wm + (size_t)t * WW;
    const float* wa = Wa + (size_t)t * WW;
    float sm = 0.0f, sa = 0.0f;
    for (int w = 0; w < WW; ++w) {
      const float a = ao[w];
      sm += wm[w] * a; sa += wa[w] * a;
    }
    multv = 1.0f / (1.0f + expf(-sm));
    addv  = fmaxf(sa + ba[t], 0.0f);
  }
  const float o  = (mDn[t] + addv) * multv;
  const float mn = tree_min(o);
  const float mx = tree_max(o);
  outv[t] = (o - mn) / (mx - mn + 1e-8f);
}

// ---------------------------------------------------------------------------
extern "C" void kernel_launch(void* const* d_in, const int* in_sizes, int n_in,
                              void* d_out, int out_size, void* d_ws, size_t ws_size,
                              hipStream_t stream) {
  (void)in_sizes; (void)n_in; (void)out_size; (void)ws_size;
  const float* x    = (const float*)d_in[0];  // (2048, 1024)
  const float* prev = (const float*)d_in[1];  // (8, 2048, 1024)
  const float* W_d  = (const float*)d_in[2];  // (1024, 1024)
  const float* in_w = (const float*)d_in[3];  // (1536, 512)
  const float* in_b = (const float*)d_in[4];  // (1536,)
  const float* ow   = (const float*)d_in[5];  // (512, 512)
  const float* ob   = (const float*)d_in[6];  // (512,)
  const float* Wm   = (const float*)d_in[7];  // (512, 512)
  const float* Wa   = (const float*)d_in[8];  // (512, 512)
  const float* ba   = (const float*)d_in[9];  // (512,)

  float* ws  = (float*)d_ws;
  float* WdT = ws;                    // 1M floats  (4 MB)
  float* S   = ws + (1u << 20);       // 1M floats  (4 MB)
  float* Bm  = ws + (2u << 20);       // 2M floats  (8 MB)
  float* Tz  = ws + (4u << 20);       // 2M floats  (8 MB)
  float* mD  = ws + (6u << 20);       // 512
  float* mDp = mD + WW;               // 8*512

  float* out = (float*)d_out;         // [0..511] = mD_out
  float* z   = out + WW;              // [512..] = z_last (2048x1024), used live

  STAR_transpose_1024<<<dim3(32, 32), dim3(32, 8), 0, stream>>>(W_d, WdT);
  // S = I - WdT @ W_d  (symmetric, so z@S^T == z@S)
  STAR_gemm_wmma<<<dim3(8, 16), 256, 0, stream>>>(WdT, W_d, S, nullptr, -1.0f, 1);
  // Bmat = x @ W_d^T = x @ WdT
  STAR_gemm_wmma<<<dim3(8, 32), 256, 0, stream>>>(x, WdT, Bm, nullptr, 1.0f, 0);
  // z0 = HT(Bmat)
  STAR_hard_thr<<<256, 256, 0, stream>>>(Bm, z);
  // 12 ISTA iterations: t = Bmat + z @ S ; z = HT(t)
  for (int it = 0; it < 12; ++it) {
    STAR_gemm_wmma<<<dim3(8, 32), 256, 0, stream>>>(z, S, Tz, Bm, 1.0f, 0);
    STAR_hard_thr<<<256, 256, 0, stream>>>(Tz, z);
  }
  // power-spectrum reductions
  STAR_zero_small<<<18, 256, 0, stream>>>(mD);
  STAR_reduce_power<<<dim3(64, 1), 256, 0, stream>>>(z, mD);
  STAR_reduce_power<<<dim3(64, NPREV), 256, 0, stream>>>(prev, mDp);
  // attention + gating epilogue -> out[0..511]
  STAR_finalize<<<1, 512, 0, stream>>>(mD, mDp, in_w, in_b, ow, ob, Wm, Wa, ba, out);
}